// ExpLogDiceLoss_91293824844156
// MI455X (gfx1250) — compile-verified
//
#include <hip/hip_runtime.h>
#include <hip/hip_bf16.h>
#include <math.h>

#define C        20
#define CPAD     64          // 60 accumulators padded to 64 (4 WMMA column groups)
#define TPB      256
#define NWAVES   (TPB / 32)
#define GAMMA    0.3f
#define LG2      0.6931471805599453f

typedef __attribute__((ext_vector_type(2))) float v2f;
typedef __attribute__((ext_vector_type(8))) float v8f;

// ---------------------------------------------------------------------------
// Workspace layout (floats): [0..19]=S_all, [20..39]=S_gt, [40..59]=ngt, pad->64
// ---------------------------------------------------------------------------
__global__ void zero_ws_kernel(float* __restrict__ ws) {
    if (threadIdx.x < CPAD) ws[threadIdx.x] = 0.0f;
}

// Issue one thread-row (80B) as 5 async b128 global->LDS transfers.
// INST_OFFSET is added to both the global and the LDS address (ISA 08 §4.4).
__device__ __forceinline__ void issue_row_async(const float* __restrict__ gsrc,
                                                const float* ldsDst) {
    unsigned lofs = (unsigned)(unsigned long long)(uintptr_t)ldsDst;  // low 32b of generic = LDS offset
    unsigned long long ga = (unsigned long long)(uintptr_t)gsrc;
    asm volatile(
        "global_load_async_to_lds_b128 %0, %1, off\n\t"
        "global_load_async_to_lds_b128 %0, %1, off offset:16\n\t"
        "global_load_async_to_lds_b128 %0, %1, off offset:32\n\t"
        "global_load_async_to_lds_b128 %0, %1, off offset:48\n\t"
        "global_load_async_to_lds_b128 %0, %1, off offset:64"
        :: "v"(lofs), "v"(ga) : "memory");
}

__global__ __launch_bounds__(TPB)
void dice_partials_kernel(const float* __restrict__ x,   // [N, 20]
                          const int*   __restrict__ lbl, // [N]
                          float*       __restrict__ ws,  // 64 floats
                          int N) {
    __shared__ float tile[2][TPB * C];          // 2 x 20KB double buffer
    __shared__ float wv[NWAVES][CPAD + 1];      // per-wave partials, +1 pad (bank stagger)

    const int tid  = threadIdx.x;
    const int lane = tid & 31;
    const int wv_i = tid >> 5;

    float accAll[C], accGt[C], accCnt[C];
#pragma unroll
    for (int c = 0; c < C; ++c) { accAll[c] = 0.f; accGt[c] = 0.f; accCnt[c] = 0.f; }

    const int tiles = (N + TPB - 1) / TPB;
    int t   = (int)blockIdx.x;
    int buf = 0;

    // Prologue: prefetch first tile (row clamped so issue count per wave is uniform).
    if (t < tiles) {
        long long row = (long long)t * TPB + tid;
        if (row >= N) row = N - 1;
        issue_row_async(x + row * (long long)C, &tile[0][tid * C]);
    }

    for (; t < tiles; t += (int)gridDim.x) {
        const int nt = t + (int)gridDim.x;
        if (nt < tiles) {                       // uniform across the wave
            long long nrow = (long long)nt * TPB + tid;
            if (nrow >= N) nrow = N - 1;
            issue_row_async(x + nrow * (long long)C, &tile[buf ^ 1][tid * C]);
            // 5 outstanding allowed (the tile just issued); current tile is complete.
            asm volatile("s_wait_asynccnt 5" ::: "memory");
        } else {
            asm volatile("s_wait_asynccnt 0" ::: "memory");
        }

        long long row   = (long long)t * TPB + tid;
        const bool valid = row < (long long)N;
        const int  lab   = valid ? lbl[row] : -1;

        const float4* r4 = (const float4*)&tile[buf][tid * C];
        float4 q0 = r4[0], q1 = r4[1], q2 = r4[2], q3 = r4[3], q4 = r4[4];
        float xr[C] = { q0.x, q0.y, q0.z, q0.w,  q1.x, q1.y, q1.z, q1.w,
                        q2.x, q2.y, q2.z, q2.w,  q3.x, q3.y, q3.z, q3.w,
                        q4.x, q4.y, q4.z, q4.w };

        float m = xr[0];
#pragma unroll
        for (int c = 1; c < C; ++c) m = fmaxf(m, xr[c]);
        float e[C]; float s = 0.f;
#pragma unroll
        for (int c = 0; c < C; ++c) { e[c] = __expf(xr[c] - m); s += e[c]; }
        const float inv = __builtin_amdgcn_rcpf(s);

        if (valid) {
#pragma unroll
            for (int c = 0; c < C; ++c) {
                const float p   = e[c] * inv;
                const float sel = (lab == c) ? 1.0f : 0.0f;
                accAll[c] += p;
                accGt[c]   = fmaf(sel, p, accGt[c]);
                accCnt[c] += sel;
            }
        }
        buf ^= 1;
    }

    // ---- wave32 butterfly reduction (lane dim) ----
#pragma unroll
    for (int c = 0; c < C; ++c) {
#pragma unroll
        for (int o = 16; o > 0; o >>= 1) {
            accAll[c] += __shfl_xor(accAll[c], o, 32);
            accGt[c]  += __shfl_xor(accGt[c],  o, 32);
            accCnt[c] += __shfl_xor(accCnt[c], o, 32);
        }
    }
    if (lane == 0) {
#pragma unroll
        for (int c = 0; c < C; ++c) {
            wv[wv_i][c]         = accAll[c];
            wv[wv_i][C + c]     = accGt[c];
            wv[wv_i][2 * C + c] = accCnt[c];
        }
#pragma unroll
        for (int c = 3 * C; c < CPAD; ++c) wv[wv_i][c] = 0.f;
    }
    __syncthreads();

    // ---- cross-wave reduction on the matrix pipe: D = ones(16x4) * B + D ----
#if __has_builtin(__builtin_amdgcn_wmma_f32_16x16x4_f32)
    if (tid < 32) {                         // wave 0, EXEC all ones (uniform branch)
        const v2f aOnes = { 1.0f, 1.0f };
        const int cl = lane & 15;           // B-matrix column N = lane%16
        const int kh = (lane >> 4) << 1;    // lanes 0-15: K={0,1}; lanes 16-31: K={2,3}
#pragma unroll
        for (int g = 0; g < 4; ++g) {       // 4 column groups of 16
            v8f d = { 0.f, 0.f, 0.f, 0.f, 0.f, 0.f, 0.f, 0.f };
#pragma unroll
            for (int st = 0; st < NWAVES / 4; ++st) {   // K tiles over 8 waves
                v2f b;
                b.x = wv[st * 4 + kh + 0][g * 16 + cl];
                b.y = wv[st * 4 + kh + 1][g * 16 + cl];
                d = __builtin_amdgcn_wmma_f32_16x16x4_f32(
                        false, aOnes, false, b, (short)0, d, false, false);
            }
            const int c = g * 16 + cl;      // row M=0 of D = column sums
            if (lane < 16 && c < 3 * C) atomicAdd(&ws[c], d[0]);
        }
    }
#else
    if (tid < 3 * C) {
        float s2 = 0.f;
        for (int w = 0; w < NWAVES; ++w) s2 += wv[w][tid];
        atomicAdd(&ws[tid], s2);
    }
#endif
}

__global__ void finalize_kernel(const float* __restrict__ ws, float* __restrict__ out) {
    if (threadIdx.x == 0 && blockIdx.x == 0) {
        float sum = 0.f; int np = 0;
        for (int c = 0; c < C; ++c) {
            const float sAll = ws[c], sGt = ws[C + c], ng = ws[2 * C + c];
            if (ng > 0.5f) {
                const float ni   = __logf(fmaxf(sGt, 1e-30f));        // log S_gt
                const float dena = __logf(ng + sAll);                  // logaddexp(log ngt, npr)
                const float ld   = LG2 + ni - dena;                    // log_dice <= 0
                sum += powf(fmaxf(-ld, 0.0f), GAMMA);
                ++np;
            }
        }
        out[0] = sum / (float)(np > 0 ? np : 1);
    }
}

extern "C" void kernel_launch(void* const* d_in, const int* in_sizes, int n_in,
                              void* d_out, int out_size, void* d_ws, size_t ws_size,
                              hipStream_t stream) {
    const float* x   = (const float*)d_in[0];
    const int*   lbl = (const int*)d_in[1];      // label (harness stages ints as int32)
    const int    N   = in_sizes[1];
    float* ws  = (float*)d_ws;
    float* out = (float*)d_out;

    zero_ws_kernel<<<1, 64, 0, stream>>>(ws);

    const int tiles = (N + TPB - 1) / TPB;
    int grid = tiles < 2048 ? tiles : 2048;
    if (grid < 1) grid = 1;
    dice_partials_kernel<<<grid, TPB, 0, stream>>>(x, lbl, ws, N);

    finalize_kernel<<<1, 32, 0, stream>>>(ws, out);
}